// LinearDiffusion_3435973837310
// MI455X (gfx1250) — compile-verified
//
#include <hip/hip_runtime.h>
#include <hip/hip_bf16.h>
#include <stdint.h>

// Problem constants (match reference)
#define N_NODES 10000
#define N_EDGES 320000
#define KH 4            // heads
#define DD 128          // feature dim
#define NF4 32          // float4 per node row (128 floats)
#define GAMMA_C 1.0f
#define DT 0.125f       // T / NSTEPS = 1/8

#define PF_DEPTH 4      // async gathers in flight per wave
#define NSLOT 8         // LDS row slots per wave (refill distance = 4 -> race-free)

// ---------------------------------------------------------------------------
// Stage coefficient tables (premultiplied by dt). pA..pD are the previously
// stored k's each stage reads; for S<=5 they are (k1,k2,k3,k4); for S==6
// they are (k1,k3,k4,k5) since b2 == 0.
// ---------------------------------------------------------------------------
template <int S> struct CF;
template <> struct CF<1> { static constexpr int NP = 0;
  static constexpr float cA = 0.f, cB = 0.f, cC = 0.f, cD = 0.f;
  static constexpr float cS = DT * 0.2f; };
template <> struct CF<2> { static constexpr int NP = 1;
  static constexpr float cA = DT * (3.0f / 40.0f), cB = 0.f, cC = 0.f, cD = 0.f;
  static constexpr float cS = DT * (9.0f / 40.0f); };
template <> struct CF<3> { static constexpr int NP = 2;
  static constexpr float cA = DT * (44.0f / 45.0f);
  static constexpr float cB = -DT * (56.0f / 15.0f);
  static constexpr float cC = 0.f, cD = 0.f;
  static constexpr float cS = DT * (32.0f / 9.0f); };
template <> struct CF<4> { static constexpr int NP = 3;
  static constexpr float cA = DT * (19372.0f / 6561.0f);
  static constexpr float cB = -DT * (25360.0f / 2187.0f);
  static constexpr float cC = DT * (64448.0f / 6561.0f);
  static constexpr float cD = 0.f;
  static constexpr float cS = -DT * (212.0f / 729.0f); };
template <> struct CF<5> { static constexpr int NP = 4;
  static constexpr float cA = DT * (9017.0f / 3168.0f);
  static constexpr float cB = -DT * (355.0f / 33.0f);
  static constexpr float cC = DT * (46732.0f / 5247.0f);
  static constexpr float cD = DT * (49.0f / 176.0f);
  static constexpr float cS = -DT * (5103.0f / 18656.0f); };
template <> struct CF<6> { static constexpr int NP = 4;
  static constexpr float cA = DT * (35.0f / 384.0f);      // k1
  static constexpr float cB = DT * (500.0f / 1113.0f);    // k3
  static constexpr float cC = DT * (125.0f / 192.0f);     // k4
  static constexpr float cD = -DT * (2187.0f / 6784.0f);  // k5
  static constexpr float cS = DT * (11.0f / 84.0f); };    // k6 (in regs)

__device__ __forceinline__ float4 fma4(float4 a, float s, float4 b) {
  a.x += s * b.x; a.y += s * b.y; a.z += s * b.z; a.w += s * b.w; return a;
}

// Stage one 512B src row into LDS: each of the 32 lanes copies 16B (b128).
__device__ __forceinline__ void async_stage_row(const float4* g, unsigned lds_off) {
  asm volatile("global_load_async_to_lds_b128 %0, %1, off"
               :: "v"(lds_off), "v"((const void*)g) : "memory");
}

// ---------------------------------------------------------------------------
// Setup kernels: zero scratch + copy h into xA
// ---------------------------------------------------------------------------
__global__ void init_kernel(const float* __restrict__ h, float* __restrict__ x,
                            float* __restrict__ e_sum, int* __restrict__ row_cnt,
                            int* __restrict__ fill_cnt) {
  const int i = blockIdx.x * blockDim.x + threadIdx.x;
  if (i < N_NODES * DD) x[i] = h[i];
  if (i < N_NODES * KH) e_sum[i] = 0.f;
  if (i < N_NODES) { row_cnt[i] = 0; fill_cnt[i] = 0; }
}

// e_sum[dst] += e  and in-degree count (one-time; atomics OK here)
__global__ void count_kernel(const float* __restrict__ e, const int* __restrict__ dst,
                             float* __restrict__ e_sum, int* __restrict__ row_cnt) {
  const int i = blockIdx.x * blockDim.x + threadIdx.x;
  if (i >= N_EDGES) return;
  const int d = dst[i];
#pragma unroll
  for (int k = 0; k < KH; ++k) atomicAdd(&e_sum[d * KH + k], e[i * KH + k]);
  atomicAdd(&row_cnt[d], 1);
}

// single-block exclusive prefix sum over N_NODES counts -> row_ptr
__global__ void scan_kernel(const int* __restrict__ cnt, int* __restrict__ ptr, int n) {
  __shared__ int tmp[1024];
  __shared__ int carry;
  const int tid = threadIdx.x;
  if (tid == 0) carry = 0;
  __syncthreads();
  for (int base = 0; base < n; base += 1024) {
    const int i = base + tid;
    const int v = (i < n) ? cnt[i] : 0;
    tmp[tid] = v;
    __syncthreads();
    for (int ofs = 1; ofs < 1024; ofs <<= 1) {
      const int t = (tid >= ofs) ? tmp[tid - ofs] : 0;
      __syncthreads();
      tmp[tid] += t;
      __syncthreads();
    }
    if (i < n) ptr[i] = carry + tmp[tid] - v;
    __syncthreads();
    if (tid == 0) carry += tmp[1023];
    __syncthreads();
  }
  if (tid == 0) ptr[n] = carry;
}

// scatter edges into dst-major CSR, with normalized weights en = e / e_sum[dst]
__global__ void fill_kernel(const float* __restrict__ e, const int* __restrict__ src,
                            const int* __restrict__ dst, const float* __restrict__ e_sum,
                            const int* __restrict__ row_ptr, int* __restrict__ fill_cnt,
                            int* __restrict__ csr_src, float* __restrict__ csr_en) {
  const int i = blockIdx.x * blockDim.x + threadIdx.x;
  if (i >= N_EDGES) return;
  const int d = dst[i];
  const int pos = row_ptr[d] + atomicAdd(&fill_cnt[d], 1);
  csr_src[pos] = src[i];
#pragma unroll
  for (int k = 0; k < KH; ++k)
    csr_en[(size_t)pos * KH + k] = e[(size_t)i * KH + k] / e_sum[d * KH + k];
}

// ---------------------------------------------------------------------------
// Fused RK stage: pull-mode SpMV, one wave32 per dst node; lane l owns
// features [4l,4l+4), head = l/8. Edge src-rows are staged into LDS with
// global_load_async_to_lds_b128 at prefetch depth 4 over 8 slots (the slot
// being refilled was consumed 4 iterations earlier -> no LDS read/write race).
// Src indices for each 32-edge chunk are fetched with ONE coalesced vector
// load and extracted on the uniform issue path with v_readlane.
// Epilogue: k_S = acc - gamma*y, then next stage input y = x + dt*sum(a_j k_j).
// ---------------------------------------------------------------------------
template <int S>
__global__ __launch_bounds__(256)
void stage_kernel(const float4* __restrict__ y_in,   // stage input (gathered)
                  const float4* __restrict__ xb,     // RK step base state
                  const float4* __restrict__ pA, const float4* __restrict__ pB,
                  const float4* __restrict__ pC, const float4* __restrict__ pD,
                  float4* __restrict__ k_store,      // k_S out (S<6)
                  float4* __restrict__ y_out,        // next stage input / x_new
                  const int* __restrict__ row_ptr,
                  const int* __restrict__ csr_src,
                  const float* __restrict__ csr_en) {
  __shared__ float4 sbuf[8][NSLOT][32];              // 8 waves x 8 slots x 512B
  const int lane = threadIdx.x & 31;
  const int wv = threadIdx.x >> 5;
  const int node = blockIdx.x * 8 + wv;
  if (node >= N_NODES) return;

  const int beg = row_ptr[node];
  const int end = row_ptr[node + 1];
  const int head = lane >> 3;
  const unsigned lds_base = (unsigned)(uintptr_t)&sbuf[wv][0][lane];

  float4 acc = make_float4(0.f, 0.f, 0.f, 0.f);

  for (int cb = beg; cb < end; cb += 32) {
    const int cn = ((end - cb) < 32) ? (end - cb) : 32;   // wave-uniform
    // one coalesced load of up to 32 src indices for this chunk
    int my_src = 0;
    if (cb + lane < end) my_src = csr_src[cb + lane];

    // prime the async pipeline
    const int npf = (cn < PF_DEPTH) ? cn : PF_DEPTH;
    for (int p = 0; p < npf; ++p) {
      const int s = __builtin_amdgcn_readlane(my_src, p);
      async_stage_row(&y_in[(size_t)s * NF4 + lane], lds_base + (unsigned)p * 512u);
    }

    for (int i = 0; i < cn; ++i) {
      // oldest of `infl` in-flight loads is done when ASYNCcnt <= infl-1
      const int infl = ((cn - i) < PF_DEPTH) ? (cn - i) : PF_DEPTH;
      if (infl >= 4)      asm volatile("s_wait_asynccnt 0x3" ::: "memory");
      else if (infl == 3) asm volatile("s_wait_asynccnt 0x2" ::: "memory");
      else if (infl == 2) asm volatile("s_wait_asynccnt 0x1" ::: "memory");
      else                asm volatile("s_wait_asynccnt 0x0" ::: "memory");

      const float w = csr_en[(size_t)(cb + i) * KH + head];
      const float4 v = sbuf[wv][i & (NSLOT - 1)][lane];
      acc.x += w * v.x; acc.y += w * v.y; acc.z += w * v.z; acc.w += w * v.w;

      if (i + PF_DEPTH < cn) {
        const int s = __builtin_amdgcn_readlane(my_src, i + PF_DEPTH);
        async_stage_row(&y_in[(size_t)s * NF4 + lane],
                        lds_base + (unsigned)((i + PF_DEPTH) & (NSLOT - 1)) * 512u);
      }
    }
  }

  const size_t idx = (size_t)node * NF4 + lane;
  const float4 yv = y_in[idx];
  float4 kv;
  kv.x = acc.x - GAMMA_C * yv.x; kv.y = acc.y - GAMMA_C * yv.y;
  kv.z = acc.z - GAMMA_C * yv.z; kv.w = acc.w - GAMMA_C * yv.w;
  if constexpr (S < 6) k_store[idx] = kv;

  float4 out = fma4(xb[idx], CF<S>::cS, kv);
  if constexpr (CF<S>::NP >= 1) out = fma4(out, CF<S>::cA, pA[idx]);
  if constexpr (CF<S>::NP >= 2) out = fma4(out, CF<S>::cB, pB[idx]);
  if constexpr (CF<S>::NP >= 3) out = fma4(out, CF<S>::cC, pC[idx]);
  if constexpr (CF<S>::NP >= 4) out = fma4(out, CF<S>::cD, pD[idx]);
  y_out[idx] = out;
}

// ---------------------------------------------------------------------------
extern "C" void kernel_launch(void* const* d_in, const int* in_sizes, int n_in,
                              void* d_out, int out_size, void* d_ws, size_t ws_size,
                              hipStream_t stream) {
  const float* h = (const float*)d_in[0];
  const float* e = (const float*)d_in[1];
  const int* src = (const int*)d_in[2];
  const int* dst = (const int*)d_in[3];

  char* ws = (char*)d_ws;
  size_t off = 0;
  auto alloc = [&](size_t bytes) -> void* {
    off = (off + 255) & ~(size_t)255;
    void* p = (void*)(ws + off);
    off += bytes;
    return p;
  };
  float* e_sum  = (float*)alloc((size_t)N_NODES * KH * 4);
  int* row_cnt  = (int*)alloc((size_t)N_NODES * 4);
  int* row_ptr  = (int*)alloc((size_t)(N_NODES + 1) * 4);
  int* fill_cnt = (int*)alloc((size_t)N_NODES * 4);
  int* csr_src  = (int*)alloc((size_t)N_EDGES * 4);
  float* csr_en = (float*)alloc((size_t)N_EDGES * KH * 4);
  const size_t row_bytes = (size_t)N_NODES * DD * 4;
  float4* xA = (float4*)alloc(row_bytes);
  float4* xB = (float4*)alloc(row_bytes);
  float4* yA = (float4*)alloc(row_bytes);
  float4* yB = (float4*)alloc(row_bytes);
  float4* k1 = (float4*)alloc(row_bytes);
  float4* k2 = (float4*)alloc(row_bytes);
  float4* k3 = (float4*)alloc(row_bytes);
  float4* k4 = (float4*)alloc(row_bytes);
  float4* k5 = (float4*)alloc(row_bytes);

  // One-time per call: zero scratch, copy h -> xA, build normalized dst-CSR.
  init_kernel<<<(N_NODES * DD + 255) / 256, 256, 0, stream>>>(h, (float*)xA, e_sum,
                                                              row_cnt, fill_cnt);
  count_kernel<<<(N_EDGES + 255) / 256, 256, 0, stream>>>(e, dst, e_sum, row_cnt);
  scan_kernel<<<1, 1024, 0, stream>>>(row_cnt, row_ptr, N_NODES);
  fill_kernel<<<(N_EDGES + 255) / 256, 256, 0, stream>>>(e, src, dst, e_sum, row_ptr,
                                                         fill_cnt, csr_src, csr_en);

  const int sgrid = (N_NODES + 7) / 8;  // 8 nodes (waves) per 256-thread block
  for (int step = 0; step < 8; ++step) {
    const float4* xin = (step & 1) ? xB : xA;
    float4* xout = (step == 7) ? (float4*)d_out : ((step & 1) ? xA : xB);
    stage_kernel<1><<<sgrid, 256, 0, stream>>>(xin, xin, nullptr, nullptr, nullptr,
                                               nullptr, k1, yA, row_ptr, csr_src, csr_en);
    stage_kernel<2><<<sgrid, 256, 0, stream>>>(yA, xin, k1, nullptr, nullptr,
                                               nullptr, k2, yB, row_ptr, csr_src, csr_en);
    stage_kernel<3><<<sgrid, 256, 0, stream>>>(yB, xin, k1, k2, nullptr,
                                               nullptr, k3, yA, row_ptr, csr_src, csr_en);
    stage_kernel<4><<<sgrid, 256, 0, stream>>>(yA, xin, k1, k2, k3,
                                               nullptr, k4, yB, row_ptr, csr_src, csr_en);
    stage_kernel<5><<<sgrid, 256, 0, stream>>>(yB, xin, k1, k2, k3,
                                               k4, k5, yA, row_ptr, csr_src, csr_en);
    stage_kernel<6><<<sgrid, 256, 0, stream>>>(yA, xin, k1, k3, k4,
                                               k5, nullptr, xout, row_ptr, csr_src, csr_en);
  }
}